// EMRouting_24369644438077
// MI455X (gfx1250) — compile-verified
//
#include <hip/hip_runtime.h>

typedef float v2f __attribute__((ext_vector_type(2)));
typedef float v8f __attribute__((ext_vector_type(8)));
typedef int   v4i __attribute__((ext_vector_type(4)));
typedef __attribute__((address_space(1))) v4i*   as1_v4i_t;
typedef __attribute__((address_space(3))) v4i*   as3_v4i_t;
typedef __attribute__((address_space(3))) float* lds_f_t;

#define N_ 32
#define I_ 2048
#define O_ 32
#define H_ 16
#define R_ 3
#define EPS_ 1e-9f
#define TI 32                   // children per LDS tile
#define TILES (I_ / TI)         // 64
#define VST 520                 // padded row stride (floats): 512 data + 8 pad
#define RRS 33                  // padded row stride for rr' tile
#define TWO_PI_ 6.2831853071795864f

#define SM_VOTES (TI * VST)     // one votes buffer (floats)
#define SM_RRP   (TI * RRS)
#define SM_TOTAL (2 * SM_VOTES + SM_RRP + 3 * O_ * H_ + 2 * O_)
#define F4_PER_TILE ((TI * O_ * H_) / 4)        // 4096
#define LD_PER_THREAD (F4_PER_TILE / 1024)      // 4

// ---- CDNA5 async global->LDS (ASYNCcnt); builtin sig: (AS1 v4i*, AS3 v4i*, imm, cpol)
#if __has_builtin(__builtin_amdgcn_global_load_async_to_lds_b128)
#define ASYNC_LD_B128(gsrc, ldst)                                          \
  __builtin_amdgcn_global_load_async_to_lds_b128(                          \
      (as1_v4i_t)(unsigned long long)(gsrc),                               \
      (as3_v4i_t)(unsigned)(unsigned long long)(ldst), 0, 0)
#else
#define ASYNC_LD_B128(gsrc, ldst)                                          \
  asm volatile("global_load_async_to_lds_b128 %0, %1, off"                 \
               :: "v"((unsigned)(unsigned long long)(ldst)),               \
                  "v"((unsigned long long)(gsrc))                          \
               : "memory")
#endif

#if __has_builtin(__builtin_amdgcn_s_wait_asynccnt)
#define WAIT_ASYNC0() __builtin_amdgcn_s_wait_asynccnt(0)
#else
#define WAIT_ASYNC0() asm volatile("s_wait_asynccnt 0x0" ::: "memory")
#endif

__global__ __launch_bounds__(1024) void em_routing_kernel(
    const float* __restrict__ votes, const float* __restrict__ activation,
    const float* __restrict__ beta_a, const float* __restrict__ beta_v,
    float* __restrict__ out)
{
  extern __shared__ float smem[];
  // AS(3)-typed base: low 32 bits of the generic LDS pointer == LDS byte offset.
  // All LDS traffic goes through this so the compiler MUST emit ds_load/ds_store
  // (round-3 asm showed generic pointers degrading to serialized flat_load).
  lds_f_t vbase = (lds_f_t)(unsigned)(unsigned long long)(smem);
  lds_f_t rrp   = vbase + 2 * SM_VOTES;   // [TI][RRS] rr' tile
  lds_f_t meanS = rrp + SM_RRP;           // [O][H]
  lds_f_t i2vS  = meanS + O_ * H_;        // [O][H]  0.5/(var+eps)
  lds_f_t lgtS  = i2vS + O_ * H_;         // [O][H]  0.5*log(2pi*(var+eps))
  lds_f_t actS  = lgtS + O_ * H_;         // [O]
  lds_f_t lactS = actS + O_;              // [O] log(act+eps)

  const int n    = blockIdx.x;
  const int t    = threadIdx.x;
  const int w    = t >> 5;                // wave id: phase-A child index / phase-B parent o
  const int lane = t & 31;                // phase-A: parent o
  const int half = lane >> 4;
  const int ls   = lane & 15;

  const float* gv = votes + (size_t)n * (I_ * O_ * H_);
  const float inv_temp_tab[3] = {0.01f * (1.0f - 0.95f),
                                 0.01f * (1.0f - 0.9025f),
                                 0.01f * (1.0f - 0.857375f)};

  for (int it = 0; it < R_; ++it) {
    v8f csv  = {0.f,0.f,0.f,0.f,0.f,0.f,0.f,0.f};   // Sv accumulator (all rows equal)
    v8f csvv = {0.f,0.f,0.f,0.f,0.f,0.f,0.f,0.f};   // Svv accumulator
    float rsum = 0.f;

    // prefetch tile 0 into buffer 0 (async, no VGPR round-trip)
    #pragma unroll
    for (int k = 0; k < LD_PER_THREAD; ++k) {
      int idx = t + k * 1024;
      int row = idx >> 7;                 // 128 float4 per child row
      int col = (idx & 127) << 2;
      ASYNC_LD_B128(gv + (size_t)row * (O_ * H_) + col, vbase + row * VST + col);
    }

    for (int tl = 0; tl < TILES; ++tl) {
      const int cur = tl & 1;
      lds_f_t vc = vbase + cur * SM_VOTES;          // arithmetic select, stays AS(3)
      WAIT_ASYNC0();                      // my slice of tile tl has landed in LDS
      __syncthreads();                    // everyone's slice has landed; prev buffers free

      // kick off tile tl+1 into the idle buffer; overlaps with E-step + WMMA below
      if (tl + 1 < TILES) {
        lds_f_t vn = vbase + (cur ^ 1) * SM_VOTES;
        const float* gn = gv + (size_t)(tl + 1) * TI * (O_ * H_);
        #pragma unroll
        for (int k = 0; k < LD_PER_THREAD; ++k) {
          int idx = t + k * 1024;
          int row = idx >> 7;
          int col = (idx & 127) << 2;
          ASYNC_LD_B128(gn + (size_t)row * (O_ * H_) + col, vn + row * VST + col);
        }
      }

      // ---- E-step: wave w = child (tl*TI + w), lane = parent o ----
      {
        float ai = activation[(size_t)n * I_ + tl * TI + w];
        float rr;
        if (it == 0) {
          rr = ai * (1.0f / O_);
        } else {
          lds_f_t vr = vc + w * VST + lane * H_;
          lds_f_t mr = meanS + lane * H_;
          lds_f_t ir = i2vS + lane * H_;
          lds_f_t gr = lgtS + lane * H_;
          float acc = 0.f;
          #pragma unroll
          for (int h = 0; h < H_; ++h) {
            float d = vr[h] - mr[h];
            acc -= d * d * ir[h] + gr[h];   // ln_p summed over pose dims
          }
          acc += lactS[lane];
          // softmax over the 32 parents == full-wave reduction
          float mx = acc;
          #pragma unroll
          for (int m = 16; m; m >>= 1) mx = fmaxf(mx, __shfl_xor(mx, m, 32));
          float e = __expf(acc - mx);
          float se = e;
          #pragma unroll
          for (int m = 16; m; m >>= 1) se += __shfl_xor(se, m, 32);
          rr = ai * e / se;
        }
        rrp[w * RRS + lane] = rr;
      }
      __syncthreads();

      // ---- M-step partials via f32 WMMA; wave w owns parent o=w ----
      // A[16x4] rows all = rr' of 4 children; B[4x16] = their pose dims (and v^2).
      #pragma unroll
      for (int c = 0; c < TI / 4; ++c) {
        int iA = 4 * c + 2 * half;          // K slots {0,1} lanes<16, {2,3} lanes>=16
        float a0 = rrp[iA * RRS + w];
        float a1 = rrp[(iA + 1) * RRS + w];
        float b0 = vc[iA * VST + w * H_ + ls];
        float b1 = vc[(iA + 1) * VST + w * H_ + ls];
        v2f Af = {a0, a1};
        v2f Bf = {b0, b1};
        v2f B2 = {b0 * b0, b1 * b1};
        csv  = __builtin_amdgcn_wmma_f32_16x16x4_f32(false, Af, false, Bf, (short)0, csv,  false, false);
        csvv = __builtin_amdgcn_wmma_f32_16x16x4_f32(false, Af, false, B2, (short)0, csvv, false, false);
        rsum += a0 + a1;                    // identical across each half's 16 lanes
      }
      // no barrier here: next tile's top barrier orders buffer reuse
    }

    // ---- finalize M-step for o = w (registers + wave shuffles only) ----
    {
      float rs   = rsum + __shfl_xor(rsum, 16);   // combine the two K-halves
      float svx  = csv[0];                        // Sv[w][h=ls]
      float svvx = csvv[0];                       // Svv[w][h=ls]
      float rse  = rs + EPS_;
      float mean = svx / rse;
      float var  = (svvx - 2.f * mean * svx + mean * mean * rs) / rse;
      var = fmaxf(var, 0.f);
      float ve   = var + EPS_;
      float cost = (beta_v[w] + 0.5f * __logf(ve)) * rs;
      float cs = cost;
      #pragma unroll
      for (int m = 8; m; m >>= 1) cs += __shfl_xor(cs, m, 16);
      float z = inv_temp_tab[it] * (beta_a[w] - cs);
      float a = 1.0f / (1.0f + __expf(-z));
      if (lane < 16) {
        meanS[w * H_ + ls] = mean;
        i2vS[w * H_ + ls]  = 0.5f / ve;
        lgtS[w * H_ + ls]  = 0.5f * __logf(TWO_PI_ * ve);
        if (ls == 0) { actS[w] = a; lactS[w] = __logf(a + EPS_); }
      }
      if (it == R_ - 1) {
        if (lane < 16) out[(size_t)n * (O_ * H_) + w * H_ + ls] = mean;    // pose
        if (lane == 0) out[(size_t)N_ * O_ * H_ + (size_t)n * O_ + w] = a; // prob
      }
    }
    __syncthreads();   // state visible before next iteration's E-step / buffer reuse
  }
}

extern "C" void kernel_launch(void* const* d_in, const int* in_sizes, int n_in,
                              void* d_out, int out_size, void* d_ws, size_t ws_size,
                              hipStream_t stream) {
  (void)in_sizes; (void)n_in; (void)out_size; (void)d_ws; (void)ws_size;
  const float* votes      = (const float*)d_in[0];
  const float* activation = (const float*)d_in[1];
  const float* beta_a     = (const float*)d_in[2];
  const float* beta_v     = (const float*)d_in[3];
  float* out = (float*)d_out;
  size_t smem_bytes = (size_t)SM_TOTAL * sizeof(float);   // ~144 KB of 320 KB WGP LDS
  hipLaunchKernelGGL(em_routing_kernel, dim3(N_), dim3(1024), smem_bytes, stream,
                     votes, activation, beta_a, beta_v, out);
}